// Decoder_83279415869594
// MI455X (gfx1250) — compile-verified
//
#include <hip/hip_runtime.h>
#include <math.h>

// ---------------- problem constants ----------------
#define BATCH 32
#define NPTS  100000
#define XDIM  256
#define IMG_ELEMS (XDIM*XDIM)          // 65536
#define PI_F 3.14159265358979323846f

typedef float v2f __attribute__((ext_vector_type(2)));
typedef float v8f __attribute__((ext_vector_type(8)));

// ---------------- pose setup: rotation rows 0/1 + shift ----------------
__global__ void pose_kernel(const float* __restrict__ alignment,
                            const float* __restrict__ shifts,
                            float* __restrict__ pose) {
    int b = threadIdx.x;
    if (b >= BATCH) return;
    float rot  = alignment[b*3+0];
    float tilt = alignment[b*3+1];
    float psi  = alignment[b*3+2];
    float ca = cosf(rot),  sa = sinf(rot);
    float cb = cosf(tilt), sb = sinf(tilt);
    float cg = cosf(psi),  sg = sinf(psi);
    float cc = cb*ca, cs = cb*sa;
    pose[b*8+0] =  cg*cc - sg*sa;
    pose[b*8+1] =  cg*cs + sg*ca;
    pose[b*8+2] = -cg*sb;
    pose[b*8+3] = -sg*cc - cg*sa;
    pose[b*8+4] = -sg*cs + cg*ca;
    pose[b*8+5] =  sg*sb;
    pose[b*8+6] = shifts[b*2+0];
    pose[b*8+7] = shifts[b*2+1];
}

// ---------------- DFT twiddle generation (exact mod-256 phase) ----------------
__global__ void twiddle_kernel(float* __restrict__ Fre,
                               float* __restrict__ Fim,
                               float* __restrict__ Fnim) {
    int idx = blockIdx.x*256 + threadIdx.x;   // 65536 threads
    int j = idx >> 8, k = idx & 255;
    int r = (j*k) & 255;                      // phase exact
    float ang = -2.0f*PI_F*(float)r*(1.0f/256.0f);
    float c = cosf(ang), s = sinf(ang);       // F = exp(-2*pi*i*j*k/N)
    Fre[idx]  = c;
    Fim[idx]  = s;                            // = -sin(2*pi*r/N)
    Fnim[idx] = -s;                           // conj(F) imaginary part
}

// ---------------- bilinear scatter: whole image in 256KB LDS ----------------
__global__ __launch_bounds__(1024)
void scatter_kernel(const float* __restrict__ pose,
                    const float* __restrict__ coords,
                    const float* __restrict__ values,
                    float* __restrict__ img) {
    extern __shared__ float simg[];           // 65536 floats = 256KB LDS
    const int b = blockIdx.x;
    const int tid = threadIdx.x;

    for (int i = tid; i < IMG_ELEMS; i += 1024) simg[i] = 0.0f;
    __syncthreads();

    const float r00 = pose[b*8+0], r01 = pose[b*8+1], r02 = pose[b*8+2];
    const float r10 = pose[b*8+3], r11 = pose[b*8+4], r12 = pose[b*8+5];
    const float sx  = pose[b*8+6], sy  = pose[b*8+7];

    for (int n = tid; n < NPTS; n += 1024) {
        float cx = coords[3*n+0], cy = coords[3*n+1], cz = coords[3*n+2];
        float px = r00*cx + r01*cy + r02*cz + sx;
        float py = r10*cx + r11*cy + r12*cz + sy;
        float gx = px + (float)(XDIM/2);
        float gy = py + (float)(XDIM/2);
        float x0f = floorf(gx), y0f = floorf(gy);
        float fx = gx - x0f,    fy = gy - y0f;
        int x0 = min(max((int)x0f, 0), XDIM-1);
        int y0 = min(max((int)y0f, 0), XDIM-1);
        int x1 = min(x0+1, XDIM-1);
        int y1 = min(y0+1, XDIM-1);
        float v = values[n];
        atomicAdd(&simg[y0*XDIM + x0], v*(1.0f-fx)*(1.0f-fy));
        atomicAdd(&simg[y0*XDIM + x1], v*fx*(1.0f-fy));
        atomicAdd(&simg[y1*XDIM + x0], v*(1.0f-fx)*fy);
        atomicAdd(&simg[y1*XDIM + x1], v*fx*fy);
    }
    __syncthreads();

    float* out = img + (size_t)b*IMG_ELEMS;
    for (int i = tid; i < IMG_ELEMS; i += 1024) out[i] = simg[i];
}

// ---------------- 7x7 'SAME' convolution, 32x32 tile + halo in LDS ----------------
__global__ __launch_bounds__(1024)
void conv7_kernel(const float* __restrict__ in,
                  float* __restrict__ out,
                  const float* __restrict__ gk) {
    __shared__ float tile[38][38];
    __shared__ float kk[49];
    const int b  = blockIdx.z;
    const int by = blockIdx.y*32, bx = blockIdx.x*32;
    const int tid = threadIdx.y*32 + threadIdx.x;
    if (tid < 49) kk[tid] = gk[tid];
    const float* src = in + (size_t)b*IMG_ELEMS;
    for (int i = tid; i < 38*38; i += 1024) {
        int ty = i / 38, tx = i % 38;
        int gy = by + ty - 3, gx = bx + tx - 3;
        float v = 0.0f;
        if (gy >= 0 && gy < XDIM && gx >= 0 && gx < XDIM)
            v = src[gy*XDIM + gx];
        tile[ty][tx] = v;
    }
    __syncthreads();
    float acc = 0.0f;
    #pragma unroll
    for (int dy = 0; dy < 7; ++dy)
        #pragma unroll
        for (int dx = 0; dx < 7; ++dx)
            acc += tile[threadIdx.y+dy][threadIdx.x+dx] * kk[dy*7+dx];
    out[(size_t)b*IMG_ELEMS + (by+threadIdx.y)*XDIM + bx + threadIdx.x] = acc;
}

// ---------------- batched complex GEMM via V_WMMA_F32_16X16X4_F32 ----------------
// C = A * B, all 256x256 row-major f32, A always complex (Are,Aim),
// B complex unless BREAL; Cim written unless !WIM. scale applied on store.
// Block = 256 threads = 8 waves, each wave owns one 16x16 C tile.
template<int BREAL, int WIM>
__global__ __launch_bounds__(256)
void cgemm256_kernel(const float* __restrict__ Are, const float* __restrict__ Aim, long sA,
                     const float* __restrict__ Bre, const float* __restrict__ Bim, long sB,
                     float* __restrict__ Cre, float* __restrict__ Cim, long sC,
                     float scale) {
    const int b    = blockIdx.y;
    const int lane = threadIdx.x & 31;
    const int tile = blockIdx.x*8 + (threadIdx.x >> 5);   // 0..255
    const int tm   = (tile >> 4) * 16;
    const int tn   = (tile & 15) * 16;

    const float* aRe = Are + (size_t)b * sA;
    const float* aIm = Aim + (size_t)b * sA;
    const float* bRe = Bre + (size_t)b * sB;
    const float* bIm = BREAL ? nullptr : (Bim + (size_t)b * sB);

    const int half = lane >> 4;       // 0: lanes 0-15, 1: lanes 16-31
    const int ln   = lane & 15;
    // A-matrix 16x4 f32 layout: lane -> row M, half-wave selects K pair {0,1}/{2,3}
    const int mA   = tm + ln;
    const int kOff = half * 2;
    // B-matrix 4x16 f32 layout: VGPR0 rows {0,1}, VGPR1 rows {2,3}; lane -> col N
    const int nB   = tn + ln;
    const int kB0  = half;
    const int kB1  = 2 + half;

    v8f cre = {0.f,0.f,0.f,0.f,0.f,0.f,0.f,0.f};
    v8f cim = cre;

    for (int kk = 0; kk < 256; kk += 4) {
        v2f are = *(const v2f*)(aRe + (size_t)mA*256 + kk + kOff);
        v2f aim = *(const v2f*)(aIm + (size_t)mA*256 + kk + kOff);
        v2f bre;
        bre.x = bRe[(size_t)(kk+kB0)*256 + nB];
        bre.y = bRe[(size_t)(kk+kB1)*256 + nB];

        // Cre += Are*Bre
        cre = __builtin_amdgcn_wmma_f32_16x16x4_f32(false, are, false, bre,
                                                    (short)0, cre, false, false);
        if (!BREAL) {
            v2f bim;
            bim.x = bIm[(size_t)(kk+kB0)*256 + nB];
            bim.y = bIm[(size_t)(kk+kB1)*256 + nB];
            v2f aimn; aimn.x = -aim.x; aimn.y = -aim.y;   // f32 WMMA has no A-neg
            // Cre -= Aim*Bim
            cre = __builtin_amdgcn_wmma_f32_16x16x4_f32(false, aimn, false, bim,
                                                        (short)0, cre, false, false);
            if (WIM) // Cim += Are*Bim
                cim = __builtin_amdgcn_wmma_f32_16x16x4_f32(false, are, false, bim,
                                                            (short)0, cim, false, false);
        }
        if (WIM)     // Cim += Aim*Bre
            cim = __builtin_amdgcn_wmma_f32_16x16x4_f32(false, aim, false, bre,
                                                        (short)0, cim, false, false);
    }

    // C/D layout: VGPR r -> row tm + r + 8*half, col = tn + ln
    float* cR = Cre + (size_t)b * sC;
    float* cI = WIM ? (Cim + (size_t)b * sC) : nullptr;
    #pragma unroll
    for (int r = 0; r < 8; ++r) {
        int row = tm + r + half*8;
        cR[(size_t)row*256 + nB] = cre[r] * scale;
        if (WIM) cI[(size_t)row*256 + nB] = cim[r] * scale;
    }
}

// ---------------- apply Hermitian-extended real CTF to G (in place) ----------------
__global__ void ctf_kernel(float* __restrict__ Gre, float* __restrict__ Gim,
                           const float* __restrict__ ctf) {
    int idx = blockIdx.x*256 + threadIdx.x;            // 32*65536 threads
    int b   = idx >> 16;
    int rem = idx & 65535;
    int k1  = rem >> 8;
    int k2  = rem & 255;
    float c;
    if (k2 <= 128) {
        c = ctf[(size_t)b*XDIM*129 + k1*129 + k2];
    } else {
        int k1m = (XDIM - k1) & 255;
        int k2m = XDIM - k2;                           // 1..127
        c = ctf[(size_t)b*XDIM*129 + k1m*129 + k2m];
    }
    Gre[idx] *= c;
    Gim[idx] *= c;
}

// ---------------- host-side orchestration ----------------
extern "C" void kernel_launch(void* const* d_in, const int* in_sizes, int n_in,
                              void* d_out, int out_size, void* d_ws, size_t ws_size,
                              hipStream_t stream) {
    const float* alignment = (const float*)d_in[0];   // (32,3)
    const float* shifts    = (const float*)d_in[1];   // (32,2)
    const float* coords    = (const float*)d_in[2];   // (100000,3)
    const float* values    = (const float*)d_in[3];   // (100000,)
    const float* gkernel   = (const float*)d_in[4];   // (7,7)
    const float* ctf       = (const float*)d_in[5];   // (32,256,129)
    float* out = (float*)d_out;                        // (32,256,256)

    float* W    = (float*)d_ws;
    const size_t IMGS = (size_t)BATCH * IMG_ELEMS;     // 2,097,152 floats
    float* W0   = W;                 // scatter img -> G.re -> junk
    float* W1   = W0 + IMGS;         // blurred img -> G.im
    float* W2   = W1 + IMGS;         // T1.re -> U.re
    float* W3   = W2 + IMGS;         // T1.im -> U.im
    float* Fre  = W3 + IMGS;
    float* Fim  = Fre + IMG_ELEMS;
    float* Fnim = Fim + IMG_ELEMS;
    float* pose = Fnim + IMG_ELEMS;  // 32*8 floats

    // 1) pose + twiddles
    pose_kernel<<<1, 32, 0, stream>>>(alignment, shifts, pose);
    twiddle_kernel<<<256, 256, 0, stream>>>(Fre, Fim, Fnim);

    // 2) bilinear scatter, one image per workgroup, 256KB dynamic LDS
    scatter_kernel<<<BATCH, 1024, IMG_ELEMS*sizeof(float), stream>>>(pose, coords, values, W0);

    // 3) 7x7 gaussian blur: W0 -> W1
    conv7_kernel<<<dim3(8,8,BATCH), dim3(32,32), 0, stream>>>(W0, W1, gkernel);

    dim3 ggrid(32, BATCH), gblk(256);
    const long S = IMG_ELEMS;

    // 4) T1 = F * img        (A = F broadcast complex, B = blurred img real)
    cgemm256_kernel<1,1><<<ggrid, gblk, 0, stream>>>(Fre, Fim, 0L,
                                                     W1, nullptr, S,
                                                     W2, W3, S, 1.0f);
    // 5) G = T1 * F          (full complex)
    cgemm256_kernel<0,1><<<ggrid, gblk, 0, stream>>>(W2, W3, S,
                                                     Fre, Fim, 0L,
                                                     W0, W1, S, 1.0f);
    // 6) H = G .* C_full(ctf)   (Hermitian-extended real filter, in place)
    ctf_kernel<<<(unsigned)(IMGS/256), 256, 0, stream>>>(W0, W1, ctf);

    // 7) U = H * conj(F)
    cgemm256_kernel<0,1><<<ggrid, gblk, 0, stream>>>(W0, W1, S,
                                                     Fre, Fnim, 0L,
                                                     W2, W3, S, 1.0f);
    // 8) out = Re( conj(F) * U ) / N^2
    cgemm256_kernel<0,0><<<ggrid, gblk, 0, stream>>>(Fre, Fnim, 0L,
                                                     W2, W3, S,
                                                     out, nullptr, S,
                                                     1.0f/65536.0f);
}